// GCN_67044439490828
// MI455X (gfx1250) — compile-verified
//
#include <hip/hip_runtime.h>
#include <hip/hip_bf16.h>

typedef __attribute__((ext_vector_type(2))) float v2f;
typedef __attribute__((ext_vector_type(8))) float v8f;

#define N_NODES_C 100000
#define F_IN  128
#define F_HID 64
#define F_OUT 32

__device__ __forceinline__ void atom_add_f32(float* p, float v) {
    unsafeAtomicAdd(p, v);  // hardware GLOBAL_ATOMIC_ADD_F32 on CDNA5
}

// ---------------- degree / normalization ----------------
__global__ void deg_kernel(const long long* __restrict__ dst,
                           float* __restrict__ deg, int E) {
    int e = blockIdx.x * blockDim.x + threadIdx.x;
    if (e < E) atom_add_f32(&deg[(int)dst[e]], 1.0f);
}

__global__ void dinv_kernel(float* __restrict__ deg, int n) {
    int i = blockIdx.x * blockDim.x + threadIdx.x;
    if (i < n) deg[i] = rsqrtf(deg[i] + 1.0f);  // self-loop => deg >= 1
}

// ---------------- GEMM via V_WMMA_F32_16X16X4_F32 ----------------
// Out[M,N] = A[M,K] @ W[K,N].  One 16x16 output tile per wave.
// A fragment: lane (half=lane>>4, m=lane&15) holds A[m][k0+2*half + {0,1}]
// B fragment: lane holds W[k0+2*half + {0,1}][colTile*16 + (lane&15)]
// C/D: vgpr v at lane -> C[v + 8*half][lane&15]
template<int K, int N, int RT, int CT>
__global__ __launch_bounds__(32 * RT * CT)
void gemm_wmma_f32(const float* __restrict__ A, const float* __restrict__ W,
                   float* __restrict__ Out, int M) {
    __shared__ float sW[K * N];
    const int tid = threadIdx.x;
    #pragma unroll 4
    for (int i = tid; i < K * N; i += 32 * RT * CT) sW[i] = W[i];
    __syncthreads();

    const int wave = tid >> 5;
    const int lane = tid & 31;
    const int half = lane >> 4;
    const int mloc = lane & 15;
    const int rt = wave / CT;
    const int ct = wave % CT;
    const int row_base = blockIdx.x * (16 * RT) + rt * 16;
    if (row_base + 16 > M) return;  // M divisible by 16*RT in this problem
    const int col = ct * 16 + mloc;

    const float* arow = A + (size_t)(row_base + mloc) * K + 2 * half;
    v8f acc = {};
    #pragma unroll
    for (int k0 = 0; k0 < K; k0 += 4) {
        v2f a = *(const v2f*)(arow + k0);
        v2f b;
        b.x = sW[(k0 + 2 * half + 0) * N + col];
        b.y = sW[(k0 + 2 * half + 1) * N + col];
        acc = __builtin_amdgcn_wmma_f32_16x16x4_f32(
            /*neg_a=*/false, a, /*neg_b=*/false, b,
            /*c_mod=*/(short)0, acc, /*reuse_a=*/false, /*reuse_b=*/false);
    }

    float* out = Out + (size_t)row_base * N + col;
    #pragma unroll
    for (int v = 0; v < 8; v++) out[(size_t)(v + 8 * half) * N] = acc[v];
}

// ---------------- edge scatter: agg[dst] += dinv[src]*dinv[dst]*h[src] ----------------
template<int H>
__global__ void scatter_kernel(const long long* __restrict__ srcs,
                               const long long* __restrict__ dsts,
                               const float* __restrict__ dinv,
                               const float* __restrict__ h,
                               float* __restrict__ agg, int E) {
    const int per_edge = H / 4;
    long long idx = (long long)blockIdx.x * blockDim.x + threadIdx.x;
    if (idx >= (long long)E * per_edge) return;
    int e  = (int)(idx / per_edge);
    int f4 = (int)(idx % per_edge) * 4;
    int s = (int)srcs[e];
    int d = (int)dsts[e];
    float ne = dinv[s] * dinv[d];
    const float4 hv = *(const float4*)(h + (size_t)s * H + f4);
    float* ap = agg + (size_t)d * H + f4;
    atom_add_f32(ap + 0, ne * hv.x);
    atom_add_f32(ap + 1, ne * hv.y);
    atom_add_f32(ap + 2, ne * hv.z);
    atom_add_f32(ap + 3, ne * hv.w);
}

// ---------------- out = [relu](agg + h*dinv^2 + bias) ----------------
template<int H, bool RELU>
__global__ void finalize_kernel(const float* __restrict__ agg,
                                const float* __restrict__ h,
                                const float* __restrict__ dinv,
                                const float* __restrict__ bias,
                                float* __restrict__ out, int n) {
    long long idx = (long long)blockIdx.x * blockDim.x + threadIdx.x;
    if (idx >= (long long)n * H) return;
    int i = (int)(idx / H);
    int f = (int)(idx % H);
    float di = dinv[i];
    float v = agg[idx] + h[idx] * di * di + bias[f];
    if (RELU) v = fmaxf(v, 0.0f);
    out[idx] = v;
}

extern "C" void kernel_launch(void* const* d_in, const int* in_sizes, int n_in,
                              void* d_out, int out_size, void* d_ws, size_t ws_size,
                              hipStream_t stream) {
    const float*     x  = (const float*)d_in[0];
    const long long* ei = (const long long*)d_in[1];   // int64 [2, E]
    const float*     W1 = (const float*)d_in[2];
    const float*     b1 = (const float*)d_in[3];
    const float*     W2 = (const float*)d_in[4];
    const float*     b2 = (const float*)d_in[5];
    float* out = (float*)d_out;

    const int N = N_NODES_C;
    const int E = in_sizes[1] / 2;
    const long long* esrc = ei;       // edge_index[0]
    const long long* edst = ei + E;   // edge_index[1]

    // workspace carve-out (256B aligned)
    char* ws = (char*)d_ws;
    size_t off = 0;
    auto carve = [&](size_t bytes) {
        void* p = ws + off;
        off += (bytes + 255) & ~(size_t)255;
        return p;
    };
    float* dinv = (float*)carve((size_t)N * 4);
    float* h1   = (float*)carve((size_t)N * F_HID * 4);
    float* agg1 = (float*)carve((size_t)N * F_HID * 4);  // becomes act1 in place
    float* h2   = (float*)carve((size_t)N * F_OUT * 4);
    float* agg2 = (float*)carve((size_t)N * F_OUT * 4);
    (void)ws_size; (void)n_in; (void)out_size;

    hipMemsetAsync(dinv, 0, (size_t)N * 4, stream);
    hipMemsetAsync(agg1, 0, (size_t)N * F_HID * 4, stream);
    hipMemsetAsync(agg2, 0, (size_t)N * F_OUT * 4, stream);

    // normalization
    deg_kernel<<<(E + 255) / 256, 256, 0, stream>>>(edst, dinv, E);
    dinv_kernel<<<(N + 255) / 256, 256, 0, stream>>>(dinv, N);

    // layer 1: h1 = x @ W1   (M=100000, K=128, N=64) ; 32 rows/block
    gemm_wmma_f32<F_IN, F_HID, 2, 4><<<N / 32, 256, 0, stream>>>(x, W1, h1, N);
    scatter_kernel<F_HID>
        <<<(int)(((long long)E * (F_HID / 4) + 255) / 256), 256, 0, stream>>>(
            esrc, edst, dinv, h1, agg1, E);
    finalize_kernel<F_HID, true>
        <<<(int)(((long long)N * F_HID + 255) / 256), 256, 0, stream>>>(
            agg1, h1, dinv, b1, agg1, N);   // relu, in place -> act1

    // layer 2: h2 = act1 @ W2 (M=100000, K=64, N=32) ; 32 rows/block
    gemm_wmma_f32<F_HID, F_OUT, 2, 2><<<N / 32, 128, 0, stream>>>(agg1, W2, h2, N);
    scatter_kernel<F_OUT>
        <<<(int)(((long long)E * (F_OUT / 4) + 255) / 256), 256, 0, stream>>>(
            esrc, edst, dinv, h2, agg2, E);
    finalize_kernel<F_OUT, false>
        <<<(int)(((long long)N * F_OUT + 255) / 256), 256, 0, stream>>>(
            agg2, h2, dinv, b2, out, N);
}